// ST2Nonlinear_3848290697509
// MI455X (gfx1250) — compile-verified
//
#include <hip/hip_runtime.h>
#include <math.h>

#define DIM   96
#define NVOX  (DIM*DIM*DIM)          // 884736
#define NFIELD 6                     // [-T0,-T1,-T2, +T1,+T2,+T3] after /128
#define NPAIR  6
#define BLOCK  256
#define NBLK   ((NVOX + BLOCK - 1)/BLOCK)   // 3456
#define NSTEPS 7

// ---------------------------------------------------------------------------
// Trilinear sample of a channels-first (3, D, H, W) fp32 volume with border
// clamp, matching the reference exactly (fx from unclipped floor; x0 clipped,
// x1 = clip(x0+1)).
// ---------------------------------------------------------------------------
__device__ __forceinline__ void trisample(const float* __restrict__ vol,
                                          float cx, float cy, float cz,
                                          float& sx, float& sy, float& sz)
{
    float x0f = floorf(cx), y0f = floorf(cy), z0f = floorf(cz);
    float fx = cx - x0f,  fy = cy - y0f,  fz = cz - z0f;
    int x0 = min(max((int)x0f, 0), DIM - 1); int x1 = min(x0 + 1, DIM - 1);
    int y0 = min(max((int)y0f, 0), DIM - 1); int y1 = min(y0 + 1, DIM - 1);
    int z0 = min(max((int)z0f, 0), DIM - 1); int z1 = min(z0 + 1, DIM - 1);
    int r00 = (x0*DIM + y0)*DIM, r01 = (x0*DIM + y1)*DIM;
    int r10 = (x1*DIM + y0)*DIM, r11 = (x1*DIM + y1)*DIM;
    int i000 = r00 + z0, i001 = r00 + z1, i010 = r01 + z0, i011 = r01 + z1;
    int i100 = r10 + z0, i101 = r10 + z1, i110 = r11 + z0, i111 = r11 + z1;
    float wz = 1.0f - fz, wy = 1.0f - fy, wx = 1.0f - fx;
    float res[3];
#pragma unroll
    for (int c = 0; c < 3; ++c) {
        const float* b = vol + c*NVOX;
        float c000 = b[i000], c001 = b[i001], c010 = b[i010], c011 = b[i011];
        float c100 = b[i100], c101 = b[i101], c110 = b[i110], c111 = b[i111];
        float c00 = c000*wz + c001*fz;
        float c01 = c010*wz + c011*fz;
        float c10 = c100*wz + c101*fz;
        float c11 = c110*wz + c111*fz;
        float c0  = c00*wy + c01*fy;
        float c1  = c10*wy + c11*fy;
        res[c] = c0*wx + c1*fx;
    }
    sx = res[0]; sy = res[1]; sz = res[2];
}

// ---------------------------------------------------------------------------
// Build the 6 scaled fields: f<3 -> -T[f]/128 ; f>=3 -> +T[f-2]/128
// ---------------------------------------------------------------------------
__global__ void k_scale(const float* __restrict__ T, float* __restrict__ out)
{
    long i = (long)blockIdx.x * blockDim.x + threadIdx.x;
    const long tot = (long)NFIELD * 3 * NVOX;
    if (i >= tot) return;
    int  f   = (int)(i / (3L * NVOX));
    long rem = i - (long)f * 3 * NVOX;
    int  t   = (f < 3) ? f : (f - 2);
    float sg = (f < 3) ? -1.0f : 1.0f;
    out[i] = sg * T[(long)t * 3 * NVOX + rem] * (1.0f / 128.0f);
}

// ---------------------------------------------------------------------------
// One scaling-and-squaring step for all 6 fields: vout = v + S(v, grid+v)
// ---------------------------------------------------------------------------
__global__ void __launch_bounds__(BLOCK)
k_step(const float* __restrict__ vin, float* __restrict__ vout)
{
    int idx = blockIdx.x * blockDim.x + threadIdx.x;
    if (idx >= NVOX) return;
    const float* v = vin  + (long)blockIdx.y * 3 * NVOX;
    float*       o = vout + (long)blockIdx.y * 3 * NVOX;
    float vx = v[idx], vy = v[NVOX + idx], vz = v[2*NVOX + idx];
    int x = idx / (DIM*DIM);
    int rem = idx - x*DIM*DIM;
    int y = rem / DIM;
    int z = rem - y*DIM;
    float sx, sy, sz;
    trisample(v, (float)x + vx, (float)y + vy, (float)z + vz, sx, sy, sz);
    o[idx]          = vx + sx;
    o[NVOX + idx]   = vy + sy;
    o[2*NVOX + idx] = vz + sz;
}

// ---------------------------------------------------------------------------
// Compose + residual norm + deterministic block reduction to partials.
// R layout: (D,H,W,3,6) -> R[idx*18 + c*6 + pair]
// ---------------------------------------------------------------------------
__global__ void __launch_bounds__(BLOCK)
k_loss(const float* __restrict__ fields, const float* __restrict__ R,
       float* __restrict__ partials)
{
    __shared__ float sm[BLOCK / 32];
    const int refs[NPAIR] = {0, 0, 0, 1, 1, 2};
    const int flos[NPAIR] = {1, 2, 3, 2, 3, 3};
    int idx  = blockIdx.x * blockDim.x + threadIdx.x;
    int pair = blockIdx.y;
    float val = 0.0f;
    if (idx < NVOX) {
        const float* fn = fields + (long)refs[pair] * 3 * NVOX;          // F_neg[ref]
        const float* fp = fields + (long)(flos[pair] + 2) * 3 * NVOX;    // F_pos[flo]
        float f1x = fn[idx], f1y = fn[NVOX + idx], f1z = fn[2*NVOX + idx];
        int x = idx / (DIM*DIM);
        int rem = idx - x*DIM*DIM;
        int y = rem / DIM;
        int z = rem - y*DIM;
        float sx, sy, sz;
        trisample(fp, (float)x + f1x, (float)y + f1y, (float)z + f1z, sx, sy, sz);
        long rb = (long)idx * 18 + pair;
        float d0 = (f1x + sx) - R[rb];
        float d1 = (f1y + sy) - R[rb + 6];
        float d2 = (f1z + sz) - R[rb + 12];
        val = sqrtf(d0*d0 + d1*d1 + d2*d2);
    }
    // wave32 shuffle reduction
#pragma unroll
    for (int off = 16; off > 0; off >>= 1) val += __shfl_down(val, off, 32);
    int lane = threadIdx.x & 31, w = threadIdx.x >> 5;
    if (lane == 0) sm[w] = val;
    __syncthreads();
    if (threadIdx.x == 0) {
        float s = 0.0f;
#pragma unroll
        for (int i = 0; i < BLOCK / 32; ++i) s += sm[i];
        partials[(long)pair * gridDim.x + blockIdx.x] = s;
    }
}

// ---------------------------------------------------------------------------
// Final deterministic reduction (single block). Per-wave reduction is done
// with V_WMMA_F32_16X16X4_F32: A = 16x4 ones, B holds 32 lane partials
// (b.y = 0), so D[m][n] = column-sums; every lane's c[0] is a column sum and
// the 32 lanes cover each of the 16 columns exactly twice -> sum * 0.5.
// ---------------------------------------------------------------------------
typedef __attribute__((ext_vector_type(2))) float v2f;
typedef __attribute__((ext_vector_type(8))) float v8f;

__global__ void __launch_bounds__(256)
k_final(const float* __restrict__ partials, int n, float* __restrict__ out)
{
    __shared__ float sm[8];
    float s = 0.0f;
    for (int i = threadIdx.x; i < n; i += 256) s += partials[i];
#if defined(__gfx1250__) && __has_builtin(__builtin_amdgcn_wmma_f32_16x16x4_f32)
    {
        v2f a; a.x = 1.0f; a.y = 1.0f;   // 16x4 all-ones A matrix
        v2f b; b.x = s;    b.y = 0.0f;   // 4x16 B matrix carrying 32 partials
        v8f c = {};
        c = __builtin_amdgcn_wmma_f32_16x16x4_f32(false, a, false, b,
                                                  (short)0, c, false, false);
        float cs = c[0];                 // column sum (each column appears twice)
#pragma unroll
        for (int off = 16; off > 0; off >>= 1) cs += __shfl_down(cs, off, 32);
        s = cs * 0.5f;
    }
#else
#pragma unroll
    for (int off = 16; off > 0; off >>= 1) s += __shfl_down(s, off, 32);
#endif
    int lane = threadIdx.x & 31, w = threadIdx.x >> 5;
    if (lane == 0) sm[w] = s;
    __syncthreads();
    if (threadIdx.x == 0) {
        float t = 0.0f;
#pragma unroll
        for (int i = 0; i < 8; ++i) t += sm[i];
        out[0] = t;
    }
}

// ---------------------------------------------------------------------------
extern "C" void kernel_launch(void* const* d_in, const int* in_sizes, int n_in,
                              void* d_out, int out_size, void* d_ws, size_t ws_size,
                              hipStream_t stream)
{
    const float* T = (const float*)d_in[0];   // (4,3,96,96,96)
    const float* R = (const float*)d_in[1];   // (96,96,96,3,6)
    float* out = (float*)d_out;               // scalar loss

    // Workspace layout: bufA | bufB | partials  (~127.6 MB total)
    float* bufA     = (float*)d_ws;
    float* bufB     = bufA + (size_t)NFIELD * 3 * NVOX;
    float* partials = bufB + (size_t)NFIELD * 3 * NVOX;

    {
        long tot = (long)NFIELD * 3 * NVOX;
        int blocks = (int)((tot + BLOCK - 1) / BLOCK);
        k_scale<<<blocks, BLOCK, 0, stream>>>(T, bufA);
    }

    float* cur = bufA;
    float* nxt = bufB;
    for (int s = 0; s < NSTEPS; ++s) {
        k_step<<<dim3(NBLK, NFIELD), BLOCK, 0, stream>>>(cur, nxt);
        float* t = cur; cur = nxt; nxt = t;
    }

    k_loss<<<dim3(NBLK, NPAIR), BLOCK, 0, stream>>>(cur, R, partials);
    k_final<<<1, 256, 0, stream>>>(partials, NPAIR * NBLK, out);
}